// torch_stft_77266461655207
// MI455X (gfx1250) — compile-verified
//
#include <hip/hip_runtime.h>

// CDNA5 / gfx1250 STFT as f32-WMMA GEMM with async-LDS staging of the DFT table.
// out[b,k,f] (complex interleaved) = sum_n T[k,n] * x[b, f*H+n],
// T folds window + ortho 1/sqrt(N) into cos/sin tables.

typedef __attribute__((ext_vector_type(2))) float v2f;
typedef __attribute__((ext_vector_type(4))) float v4f;
typedef __attribute__((ext_vector_type(8))) float v8f;

#define FFT_N   1024
#define HOP     256
#define NBATCH  32
#define TLEN    262144
#define NFRAMES 1021     // (T - N)/H + 1
#define NBINS   513      // N/2 + 1
#define KPAD    528      // NBINS padded to multiple of 16
#define KTILES  33       // KPAD / 16
#define TSTEPS  256      // FFT_N / 4 K-steps
#define ROWF    2048     // floats per k-row of combined table (re+im)

// ---------------------------------------------------------------------------
// Combined DFT table, laid out for single-b128 A-operand fetches:
//   tabC[k][h][t][4] = { re(n), re(n+1), im(n), im(n+1) },  n = 4t + 2h
// where h = lane>>4 selects the WMMA half, t = K-step. Rows k >= NBINS zero.
// Size: KPAD * 2048 floats = 4.33 MB of d_ws.
// ---------------------------------------------------------------------------
__global__ void build_dft_tables(const float* __restrict__ w,
                                 float* __restrict__ tabC) {
    int idx = blockIdx.x * blockDim.x + threadIdx.x;  // one (k,h,t) per thread
    if (idx >= KPAD * 2 * TSTEPS) return;
    int t = idx & (TSTEPS - 1);
    int h = (idx >> 8) & 1;
    int k = idx >> 9;
    v4f val = {0.0f, 0.0f, 0.0f, 0.0f};
    if (k < NBINS) {
        int n0 = 4 * t + 2 * h;
        #pragma unroll
        for (int j = 0; j < 2; ++j) {
            int n  = n0 + j;
            int kn = (k * n) & (FFT_N - 1);   // exact modular reduction
            float theta = -6.283185307179586f * (float)kn * (1.0f / (float)FFT_N);
            float s, c;
            __sincosf(theta, &s, &c);
            float sc = w[n] * 0.03125f;       // window * 1/sqrt(1024)
            val[j]     = c * sc;
            val[2 + j] = s * sc;
        }
    }
    *(v4f*)(tabC + (size_t)idx * 4) = val;
}

// ---------------------------------------------------------------------------
// Main GEMM. Block = 16 k-bins x 256 frames (8 waves x 2 frame-tiles).
// A (16x4 f32): from LDS-staged table, one ds_load_b128 yields re+im pairs.
// B (4x16 f32): raw signal, clamped addresses for pad frames (garbage only
//               lands in never-stored output columns -> no EXEC divergence).
// ---------------------------------------------------------------------------
__global__ __launch_bounds__(256)
void stft_wmma_kernel(const float* __restrict__ x,
                      const float* __restrict__ tabC,
                      float* __restrict__ out) {
    __shared__ float sA[16 * ROWF];   // 128 KB: k-tile's combined table slice

    const int lane = threadIdx.x & 31;
    const int wave = threadIdx.x >> 5;
    const int half = lane >> 4;       // WMMA half: lanes 0-15 / 16-31
    const int l15  = lane & 15;

    const int k0 = blockIdx.x * 16;
    const int b  = blockIdx.z;

    // ---- Stage 16 k-rows of tabC into LDS with gfx1250 async copies ----
    {
        unsigned lds_base = (unsigned)(unsigned long long)(const void*)sA;
        unsigned long long gbase =
            (unsigned long long)(tabC + (size_t)k0 * ROWF);
        // 128 KB / 256 threads = 32 x b128 per thread
        #pragma unroll
        for (int i = 0; i < 32; ++i) {
            unsigned e = (unsigned)threadIdx.x + (unsigned)i * 256u;  // float4 idx
            unsigned laddr = lds_base + e * 16u;
            unsigned long long gaddr = gbase + (unsigned long long)e * 16ull;
            asm volatile("global_load_async_to_lds_b128 %0, %1, off"
                         :: "v"(laddr), "v"(gaddr) : "memory");
        }
        asm volatile("s_wait_asynccnt 0x0" ::: "memory");
    }
    __syncthreads();

    // ---- Per-wave operand pointers ----
    // A from LDS: lane (l15, half) stream is contiguous across K-steps.
    const v4f* aptr = (const v4f*)(sA + (size_t)(l15 * 2 + half) * (TSTEPS * 4));

    // B from global: two frame tiles per wave; clamp pad frames.
    const int ft0 = (blockIdx.y * 8 + wave) * 32;        // first frame of tile 0
    const int f_0 = ft0 + l15;                           // tile-0 frame (lane)
    const int f_1 = ft0 + 16 + l15;                      // tile-1 frame (lane)
    const int fc0 = (f_0 < NFRAMES) ? f_0 : (NFRAMES - 1);
    const int fc1 = (f_1 < NFRAMES) ? f_1 : (NFRAMES - 1);
    const float* xbase = x + (size_t)b * TLEN + 2 * half;
    const float* xb0 = xbase + (size_t)fc0 * HOP;
    const float* xb1 = xbase + (size_t)fc1 * HOP;

    v8f acc_re0 = {}, acc_im0 = {};
    v8f acc_re1 = {}, acc_im1 = {};

    #pragma unroll 4
    for (int t = 0; t < TSTEPS; ++t) {
        v4f a  = aptr[t];                         // ds_load_b128: re pair + im pair
        v2f b0 = *(const v2f*)(xb0 + 4 * t);      // global_load_b64
        v2f b1 = *(const v2f*)(xb1 + 4 * t);
        v2f are = {a[0], a[1]};
        v2f aim = {a[2], a[3]};

        // (neg_a, A, neg_b, B, c_mod, C, reuse_a, reuse_b)
        acc_re0 = __builtin_amdgcn_wmma_f32_16x16x4_f32(
            false, are, false, b0, (short)0, acc_re0, false, false);
        acc_im0 = __builtin_amdgcn_wmma_f32_16x16x4_f32(
            false, aim, false, b0, (short)0, acc_im0, false, false);
        acc_re1 = __builtin_amdgcn_wmma_f32_16x16x4_f32(
            false, are, false, b1, (short)0, acc_re1, false, false);
        acc_im1 = __builtin_amdgcn_wmma_f32_16x16x4_f32(
            false, aim, false, b1, (short)0, acc_im1, false, false);
    }

    // ---- Store interleaved complex out[b][k][f] (b64 per element pair) ----
    #pragma unroll
    for (int r = 0; r < 8; ++r) {
        int k = k0 + r + 8 * half;
        if (k < NBINS) {
            size_t rowoff = ((size_t)b * NBINS + k) * NFRAMES;
            if (f_0 < NFRAMES) {
                v2f c = {acc_re0[r], acc_im0[r]};
                *(v2f*)(out + (rowoff + f_0) * 2) = c;
            }
            if (f_1 < NFRAMES) {
                v2f c = {acc_re1[r], acc_im1[r]};
                *(v2f*)(out + (rowoff + f_1) * 2) = c;
            }
        }
    }
}

extern "C" void kernel_launch(void* const* d_in, const int* in_sizes, int n_in,
                              void* d_out, int out_size, void* d_ws, size_t ws_size,
                              hipStream_t stream) {
    const float* x = (const float*)d_in[0];   // [B, T] float32
    const float* w = (const float*)d_in[1];   // [N]    float32
    float* out = (float*)d_out;               // [B, 513, 1021] complex -> float pairs

    float* tabC = (float*)d_ws;               // 4.33 MB combined DFT table

    const int tab_thr = KPAD * 2 * TSTEPS;    // one thread per (k,h,t)
    build_dft_tables<<<(tab_thr + 255) / 256, 256, 0, stream>>>(w, tabC);

    dim3 grid(KTILES, 4, NBATCH);  // 33 k-tiles x (4*8 waves*32 frames=1024) x 32
    stft_wmma_kernel<<<grid, 256, 0, stream>>>(x, tabC, out);
}